// GraphDecoder_30949534335630
// MI455X (gfx1250) — compile-verified
//
#include <hip/hip_runtime.h>
#include <hip/hip_bf16.h>

typedef __attribute__((ext_vector_type(2))) float v2f;
typedef __attribute__((ext_vector_type(4))) float v4f;
typedef __attribute__((ext_vector_type(8))) float v8f;

// ---------------------------------------------------------------------------
// CSR construction: count -> single-block scan -> scatter
// ---------------------------------------------------------------------------
__global__ void k_fill_i32(int* __restrict__ p, int v, int n) {
    int i = blockIdx.x * blockDim.x + threadIdx.x;
    if (i < n) p[i] = v;
}

__global__ void k_count(const int* __restrict__ dst, int* __restrict__ cnt, int E) {
    int e = blockIdx.x * blockDim.x + threadIdx.x;
    if (e < E) atomicAdd(&cnt[dst[e]], 1);
}

#define SCAN_T 1024
#define SCAN_C 10
__global__ __launch_bounds__(SCAN_T)
void k_scan(int* __restrict__ cnt, int* __restrict__ rowptr, int NN) {
    __shared__ int lds[SCAN_T];
    int t = threadIdx.x;
    int base = t * SCAN_C;
    int local[SCAN_C];
    int s = 0;
#pragma unroll
    for (int e = 0; e < SCAN_C; ++e) {
        int idx = base + e;
        int v = (idx < NN) ? cnt[idx] : 0;
        local[e] = v;
        s += v;
    }
    lds[t] = s;
    __syncthreads();
    for (int off = 1; off < SCAN_T; off <<= 1) {
        int v = (t >= off) ? lds[t - off] : 0;
        __syncthreads();
        lds[t] += v;
        __syncthreads();
    }
    int run = lds[t] - s;
#pragma unroll
    for (int e = 0; e < SCAN_C; ++e) {
        int idx = base + e;
        if (idx < NN) {
            rowptr[idx] = run;
            cnt[idx]    = run;
            run += local[e];
        }
    }
    if (t == 0) rowptr[NN] = lds[SCAN_T - 1];
}

__global__ void k_scatter(const int* __restrict__ src, const int* __restrict__ dst,
                          int* __restrict__ cursor, int* __restrict__ csr, int E) {
    int e = blockIdx.x * blockDim.x + threadIdx.x;
    if (e < E) {
        int p = atomicAdd(&cursor[dst[e]], 1);
        csr[p] = src[e];
    }
}

// ---------------------------------------------------------------------------
// Mean aggregation: one 2-wave block (64 thr) per node, float4 row gathers.
// All feature matrices (<= 20.5 MB) sit in the 192 MB L2, so this is an
// L2-rate gather: 512 B contiguous per wave per neighbor row.
// d is 256 or 512.
// ---------------------------------------------------------------------------
__global__ __launch_bounds__(64)
void k_aggregate(const float* __restrict__ x, int d,
                 const int* __restrict__ rowptr, const int* __restrict__ csr,
                 float* __restrict__ mean) {
    const int node = blockIdx.x;
    const int beg = rowptr[node];
    const int end = rowptr[node + 1];
    const float inv = 1.0f / fmaxf((float)(end - beg), 1.0f);
    const int t = threadIdx.x;
    const int dq = d >> 2;                 // float4s per row: 64 or 128
    const v4f* __restrict__ x4 = (const v4f*)x;

    v4f a0 = {}, a1 = {};
    int j = beg;
    for (; j + 1 < end; j += 2) {
        int s0 = csr[j];
        int s1 = csr[j + 1];
        const v4f* r0 = x4 + s0 * dq;
        const v4f* r1 = x4 + s1 * dq;
        a0 += r0[t];
        a0 += r1[t];
        if (d > 256) { a1 += r0[t + 64]; a1 += r1[t + 64]; }
    }
    if (j < end) {
        const v4f* r0 = x4 + csr[j] * dq;
        a0 += r0[t];
        if (d > 256) a1 += r0[t + 64];
    }
    v4f* m4 = (v4f*)mean + node * dq;
    m4[t] = a0 * inv;
    if (d > 256) m4[t + 64] = a1 * inv;
}

// ---------------------------------------------------------------------------
// Fused SAGE linear: out = mean @ Wl + xr @ Wr + b, optional ReLU.
// fp32 WMMA 16x16x4 (exact fp32 math). Block tile 64x64, 8 waves; each wave
// owns a 16x32 strip (two v8f accumulators sharing one A fragment).
// ---------------------------------------------------------------------------
__device__ __forceinline__ void sage_phase(
    const float* __restrict__ A, const float* __restrict__ W,
    int M, int K, int N, int m0, int n0,
    int tid, int mSub, int nSub, int lhalf, int koff,
    float (*__restrict__ As)[65], float (*__restrict__ Bs)[72],
    v8f& acc0, v8f& acc1) {
    for (int k0 = 0; k0 < K; k0 += 32) {
        // --- Stage A block (64 rows x 32 k) transposed into As[k][m] ---
        {
            const int k4 = (tid & 7) * 4;    // 0,4,...,28
            const int i  = tid >> 3;         // 0..31
#pragma unroll
            for (int p = 0; p < 2; ++p) {
                int ii = i + p * 32;
                int row = m0 + ii;
                int rowc = row < M ? row : M - 1;   // clamp: pad rows never stored
                v4f v = *(const v4f*)(A + rowc * K + k0 + k4);
                As[k4 + 0][ii] = v.x;
                As[k4 + 1][ii] = v.y;
                As[k4 + 2][ii] = v.z;
                As[k4 + 3][ii] = v.w;
            }
        }
        // --- Stage B block (32 k x 64 n) into Bs[k][n] via b128 ---
        {
            const int n4 = (tid & 15) * 4;   // 0,4,...,60
            const int k  = tid >> 4;         // 0..15
#pragma unroll
            for (int p = 0; p < 2; ++p) {
                int kk2 = k + p * 16;
                *(v4f*)&Bs[kk2][n4] = *(const v4f*)(W + (k0 + kk2) * N + n0 + n4);
            }
        }
        __syncthreads();
#pragma unroll
        for (int kk = 0; kk < 32; kk += 4) {
            v2f a, b0, b1;
            a.x  = As[kk + koff][mSub + lhalf];
            a.y  = As[kk + koff + 1][mSub + lhalf];
            b0.x = Bs[kk + koff][nSub + lhalf];
            b0.y = Bs[kk + koff + 1][nSub + lhalf];
            b1.x = Bs[kk + koff][nSub + lhalf + 16];
            b1.y = Bs[kk + koff + 1][nSub + lhalf + 16];
            acc0 = __builtin_amdgcn_wmma_f32_16x16x4_f32(
                false, a, false, b0, (short)0, acc0, false, false);
            acc1 = __builtin_amdgcn_wmma_f32_16x16x4_f32(
                false, a, false, b1, (short)0, acc1, false, false);
        }
        __syncthreads();
    }
}

__global__ __launch_bounds__(256)
void k_sage_gemm(const float* __restrict__ Amean, const float* __restrict__ Aroot,
                 const float* __restrict__ Wl, const float* __restrict__ Wr,
                 const float* __restrict__ bias, float* __restrict__ out,
                 int M, int K, int N, int relu) {
    __shared__ __align__(16) float As[32][65];   // stride 65 = 1 mod 64 banks
    __shared__ __align__(16) float Bs[32][72];   // stride 72: b128-aligned, halves disjoint

    const int tid  = threadIdx.x;
    const int lane = tid & 31;
    const int wave = tid >> 5;
    const int m0 = blockIdx.y * 64;
    const int n0 = blockIdx.x * 64;
    const int mSub = (wave & 3) * 16;
    const int nSub = (wave >> 2) * 32;
    const int lhalf = lane & 15;
    const int koff  = (lane >> 4) * 2;

    v8f acc0 = {}, acc1 = {};

    sage_phase(Amean, Wl, M, K, N, m0, n0, tid, mSub, nSub, lhalf, koff, As, Bs, acc0, acc1);
    sage_phase(Aroot, Wr, M, K, N, m0, n0, tid, mSub, nSub, lhalf, koff, As, Bs, acc0, acc1);

    const int col = n0 + nSub + lhalf;
    const float bv0 = bias[col];
    const float bv1 = bias[col + 16];
    const int rbase = m0 + mSub + (lane >> 4) * 8;
#pragma unroll
    for (int r = 0; r < 8; ++r) {
        int row = rbase + r;
        if (row < M) {
            float v0 = acc0[r] + bv0;
            float v1 = acc1[r] + bv1;
            if (relu) { v0 = fmaxf(v0, 0.0f); v1 = fmaxf(v1, 0.0f); }
            out[row * N + col]      = v0;
            out[row * N + col + 16] = v1;
        }
    }
}

// ---------------------------------------------------------------------------
// Driver
// ---------------------------------------------------------------------------
static inline size_t alignup(size_t v) { return (v + 255) & ~(size_t)255; }

extern "C" void kernel_launch(void* const* d_in, const int* in_sizes, int n_in,
                              void* d_out, int out_size, void* d_ws, size_t ws_size,
                              hipStream_t stream) {
    const float* z    = (const float*)d_in[0];
    const int*   edge = (const int*)d_in[1];
    const float* Wl1 = (const float*)d_in[2];
    const float* Wr1 = (const float*)d_in[3];
    const float* b1  = (const float*)d_in[4];
    const float* Wl2 = (const float*)d_in[5];
    const float* Wr2 = (const float*)d_in[6];
    const float* b2  = (const float*)d_in[7];
    const float* Wl3 = (const float*)d_in[8];
    const float* Wr3 = (const float*)d_in[9];
    const float* b3  = (const float*)d_in[10];
    const float* Wl4 = (const float*)d_in[11];
    const float* Wr4 = (const float*)d_in[12];
    const float* b4  = (const float*)d_in[13];

    const int NN = in_sizes[0] / 256;   // nodes (z is NN x 256)
    const int E  = in_sizes[1] / 2;     // edges (edge_index is 2 x E)
    const int* src = edge;
    const int* dst = edge + E;

    char* w = (char*)d_ws;
    int* rowptr = (int*)w;   w += alignup(sizeof(int) * (size_t)(NN + 1));
    int* cursor = (int*)w;   w += alignup(sizeof(int) * (size_t)NN);
    int* csr    = (int*)w;   w += alignup(sizeof(int) * (size_t)E);
    float* mean = (float*)w; w += alignup(sizeof(float) * (size_t)NN * 512);
    float* hA   = (float*)w; w += alignup(sizeof(float) * (size_t)NN * 512);
    float* hB   = (float*)w;

    const int T = 256;
    // ---- CSR build (reused by all 4 layers) ----
    k_fill_i32<<<(NN + T - 1) / T, T, 0, stream>>>(cursor, 0, NN);
    k_count<<<(E + T - 1) / T, T, 0, stream>>>(dst, cursor, E);
    k_scan<<<1, SCAN_T, 0, stream>>>(cursor, rowptr, NN);
    k_scatter<<<(E + T - 1) / T, T, 0, stream>>>(src, dst, cursor, csr, E);

    float* outf = (float*)d_out;
    const int MB = (NN + 63) / 64;

    // Layer 1: 256 -> 512, ReLU
    k_aggregate<<<NN, 64, 0, stream>>>(z, 256, rowptr, csr, mean);
    k_sage_gemm<<<dim3(512 / 64, MB), T, 0, stream>>>(
        mean, z, Wl1, Wr1, b1, hA, NN, 256, 512, 1);

    // Layer 2: 512 -> 512, ReLU
    k_aggregate<<<NN, 64, 0, stream>>>(hA, 512, rowptr, csr, mean);
    k_sage_gemm<<<dim3(512 / 64, MB), T, 0, stream>>>(
        mean, hA, Wl2, Wr2, b2, hB, NN, 512, 512, 1);

    // Layer 3: 512 -> 256, ReLU
    k_aggregate<<<NN, 64, 0, stream>>>(hB, 512, rowptr, csr, mean);
    k_sage_gemm<<<dim3(256 / 64, MB), T, 0, stream>>>(
        mean, hB, Wl3, Wr3, b3, hA, NN, 512, 256, 1);

    // Layer 4: 256 -> 512, no ReLU
    k_aggregate<<<NN, 64, 0, stream>>>(hA, 256, rowptr, csr, mean);
    k_sage_gemm<<<dim3(512 / 64, MB), T, 0, stream>>>(
        mean, hA, Wl4, Wr4, b4, outf, NN, 256, 512, 0);
}